// DWMCell_4896262718021
// MI455X (gfx1250) — compile-verified
//
#include <hip/hip_runtime.h>
#include <math.h>

#define B_      64
#define IN_DIM  64
#define STATE   512
#define H_      4
#define M_      32
#define N_      8192
#define OUT_    64
#define READ_   (H_ * M_)          // 128
#define PER_HEAD 72                // 3 + 1 + 3 + 1 + 32 + 32
#define UPDATE_ (H_ * PER_HEAD)    // 288
#define COMB_   (IN_DIM + STATE + READ_)  // 704
#define EPSF    1e-12f

typedef __attribute__((ext_vector_type(2))) float v2f;
typedef __attribute__((ext_vector_type(8))) float v8f;

// ---------------------------------------------------------------------------
// Phase 0: pack x (cols 0..63) and hidden_state (cols 64..575) into comb_ws.
// ---------------------------------------------------------------------------
__global__ void __launch_bounds__(256) pack_kernel(
    const float* __restrict__ x, const float* __restrict__ hid,
    float* __restrict__ comb_ws) {
  const int b = blockIdx.x;
  const int tid = threadIdx.x;
  float* row = comb_ws + (size_t)b * COMB_;
  for (int c = tid; c < IN_DIM + STATE; c += 256) {
    row[c] = (c < IN_DIM) ? x[b * IN_DIM + c] : hid[b * STATE + (c - IN_DIM)];
  }
}

// ---------------------------------------------------------------------------
// Phase 1: read[b,h,m] = sum_n mem[b,m,n] * head_weight[b,h,n]
// One block per (b,m); float4 streaming of the mem row (HBM-bound).
// Results written straight into comb_ws cols 576 + h*M + m.
// ---------------------------------------------------------------------------
__global__ void __launch_bounds__(256) read_kernel(
    const float* __restrict__ mem, const float* __restrict__ hw,
    float* __restrict__ comb_ws) {
  const int b = blockIdx.x / M_;
  const int m = blockIdx.x % M_;
  const int tid = threadIdx.x;

  const float4* mrow = (const float4*)(mem + ((size_t)b * M_ + m) * N_);
  const float4* w0 = (const float4*)(hw + ((size_t)b * H_ + 0) * N_);
  const float4* w1 = w0 + (N_ / 4);
  const float4* w2 = w1 + (N_ / 4);
  const float4* w3 = w2 + (N_ / 4);

  float a0 = 0.f, a1 = 0.f, a2 = 0.f, a3 = 0.f;
  for (int i = tid; i < N_ / 4; i += 256) {
    __builtin_prefetch(&mrow[i + 512], 0, 0);   // look ahead 2 strides
    float4 mv = mrow[i];
    float4 v;
    v = w0[i]; a0 += mv.x * v.x + mv.y * v.y + mv.z * v.z + mv.w * v.w;
    v = w1[i]; a1 += mv.x * v.x + mv.y * v.y + mv.z * v.z + mv.w * v.w;
    v = w2[i]; a2 += mv.x * v.x + mv.y * v.y + mv.z * v.z + mv.w * v.w;
    v = w3[i]; a3 += mv.x * v.x + mv.y * v.y + mv.z * v.z + mv.w * v.w;
  }

  __shared__ float red[4][256];
  red[0][tid] = a0; red[1][tid] = a1; red[2][tid] = a2; red[3][tid] = a3;
  __syncthreads();
  for (int s = 128; s > 0; s >>= 1) {
    if (tid < s) {
      red[0][tid] += red[0][tid + s];
      red[1][tid] += red[1][tid + s];
      red[2][tid] += red[2][tid + s];
      red[3][tid] += red[3][tid + s];
    }
    __syncthreads();
  }
  if (tid == 0) {
    float* dst = comb_ws + (size_t)b * COMB_ + (IN_DIM + STATE);
    dst[0 * M_ + m] = red[0][0];
    dst[1 * M_ + m] = red[1][0];
    dst[2 * M_ + m] = red[2][0];
    dst[3 * M_ + m] = red[3][0];
  }
}

// ---------------------------------------------------------------------------
// Phase 2: controller GEMM with V_WMMA_F32_16X16X4_F32, templated on the
// weight leading dimension so all B addresses fold to immediate offsets.
//   REGION 0: W_h (LD=512) -> h_new = sigmoid(z)
//   REGION 1: W_y (LD=64)  -> y = z
//   REGION 2: W_u (LD=288) -> u (raw, to workspace)
// One wave per 16x16 output tile.
// A fragment (16x4 f32): lanes 0-15 hold K=0,1; lanes 16-31 hold K=2,3.
// B fragment (4x16 f32): same K split; column = lane & 15.
// ---------------------------------------------------------------------------
template <int LD, int REGION>
__global__ void __launch_bounds__(32) controller_kernel(
    const float* __restrict__ comb_ws,
    const float* __restrict__ W, const float* __restrict__ bias,
    float* __restrict__ outp) {
  const int rowTile = blockIdx.x;        // 0..3  (batch tiles of 16)
  const int colBase = blockIdx.y * 16;   // column tile within this region
  const int lane = threadIdx.x;

  const int r15 = lane & 15;
  const int row = rowTile * 16 + r15;    // batch index for A fragment
  const int colLocal = colBase + r15;    // column for B fragment
  const int khalf = (lane >> 4) << 1;    // 0 for lanes 0-15, 2 for lanes 16-31

  // A: pairs (comb[row, ks], comb[row, ks+1]) with ks even -> float2 loads
  const v2f* Arow = (const v2f*)(comb_ws + (size_t)row * COMB_) + (khalf >> 1);
  // B: W[(k+khalf)*LD + colLocal] with compile-time LD -> immediate offsets
  const float* Bcol = W + (size_t)khalf * LD + colLocal;

  v8f acc = {};
#pragma unroll 8
  for (int k = 0; k < COMB_; k += 4) {
    v2f a = Arow[k >> 1];
    v2f bf;
    bf.x = Bcol[(size_t)k * LD];
    bf.y = Bcol[(size_t)(k + 1) * LD];
    acc = __builtin_amdgcn_wmma_f32_16x16x4_f32(
        /*neg_a=*/false, a, /*neg_b=*/false, bf,
        /*c_mod=*/(short)0, acc, /*reuse_a=*/false, /*reuse_b=*/false);
  }

  const float bv = bias[colLocal];
  const int rbase = rowTile * 16 + ((lane < 16) ? 0 : 8);
#pragma unroll
  for (int r = 0; r < 8; ++r) {
    const int rowc = rbase + r;
    const float z = acc[r] + bv;
    if (REGION == 0)      outp[rowc * STATE + colLocal]   = 1.f / (1.f + __expf(-z));
    else if (REGION == 1) outp[rowc * OUT_ + colLocal]    = z;
    else                  outp[rowc * UPDATE_ + colLocal] = z;
  }
}

// ---------------------------------------------------------------------------
// Phase 3: addressing. One block per (b,h). Computes snapshot update,
// circular conv (3 taps), sharpen (pow gamma), renormalize over N=8192.
// Sharpened values held in 32KB LDS between sum and normalize passes.
// Also parses e (sigmoid) and a into workspace for phase 4.
// ---------------------------------------------------------------------------
__device__ __forceinline__ float softplus_f(float v) {
  return (v > 20.f) ? v : log1pf(__expf(v));
}

__global__ void __launch_bounds__(256) address_kernel(
    const float* __restrict__ hw, const float* __restrict__ sw,
    const float* __restrict__ u_ws,
    float* __restrict__ wt_out, float* __restrict__ snap_out,
    float* __restrict__ e_ws, float* __restrict__ a_ws) {
  const int b = blockIdx.x / H_;
  const int h = blockIdx.x % H_;
  const int tid = threadIdx.x;
  const float* u = u_ws + (b * UPDATE_ + h * PER_HEAD);

  // Scalars (computed redundantly by all threads; same addresses -> cheap)
  const float sp0 = softplus_f(u[0]), sp1 = softplus_f(u[1]), sp2 = softplus_f(u[2]);
  const float smx = fmaxf(sp0, fmaxf(sp1, sp2));
  float se0 = __expf(sp0 - smx), se1 = __expf(sp1 - smx), se2 = __expf(sp2 - smx);
  const float ses = se0 + se1 + se2;
  const float s0 = se0 / ses, s1 = se1 / ses, s2 = se2 / ses;

  const float jd = 1.f / (1.f + __expf(-u[3]));

  const float jmx = fmaxf(u[4], fmaxf(u[5], u[6]));
  float je0 = __expf(u[4] - jmx), je1 = __expf(u[5] - jmx), je2 = __expf(u[6] - jmx);
  const float jes = je0 + je1 + je2;
  const float j0 = je0 / jes, j1 = je1 / jes, j2 = je2 / jes;

  const float gamma = 1.f + softplus_f(u[7]);

  if (tid < M_) {
    e_ws[(b * H_ + h) * M_ + tid] = 1.f / (1.f + __expf(-u[8 + tid]));
    a_ws[(b * H_ + h) * M_ + tid] = u[8 + M_ + tid];
  }

  const float* hwp = hw + (size_t)(b * H_ + h) * N_;
  const float* swp = sw + (size_t)(b * H_ + h) * N_;
  float* snp = snap_out + (size_t)(b * H_ + h) * N_;
  float* wtp = wt_out + (size_t)(b * H_ + h) * N_;

  __shared__ float sharp[N_];
  __shared__ float red[256];

  // pre-conv weight as a pure function of index (neighbors recomputed; cached)
  auto wpre = [&](int n) -> float {
    const float hv = hwp[n], sv = swp[n];
    const float sn = (1.f - jd) * sv + jd * hv;
    float w = j0 * hv + j1 * sn;
    if (n == 0) w += j2;
    return w;
  };

  float local = 0.f;
  for (int n = tid; n < N_; n += 256) {
    const float hv = hwp[n], sv = swp[n];
    const float sn = (1.f - jd) * sv + jd * hv;
    snp[n] = sn;
    float wc = j0 * hv + j1 * sn;
    if (n == 0) wc += j2;
    const int np = (n + 1) & (N_ - 1);
    const int nm = (n - 1) & (N_ - 1);
    // roll(w, i-1): tap0 = w[n+1], tap1 = w[n], tap2 = w[n-1]
    const float conv = s0 * wpre(np) + s1 * wc + s2 * wpre(nm);
    const float sh = powf(conv + EPSF, gamma);
    sharp[n] = sh;
    local += sh;
  }
  red[tid] = local;
  __syncthreads();
  for (int s = 128; s > 0; s >>= 1) {
    if (tid < s) red[tid] += red[tid + s];
    __syncthreads();
  }
  const float inv = 1.f / (red[0] + EPSF);
  for (int n = tid; n < N_; n += 256) wtp[n] = sharp[n] * inv;
}

// ---------------------------------------------------------------------------
// Phase 4: mem_new[b,m,n] = mem * prod_h(1 - e[b,h,m]*wt[b,h,n])
//                           + sum_h a[b,h,m]*wt[b,h,n]
// One block per (b,m); float4 streaming (HBM-bound: 67MB read + 67MB write).
// ---------------------------------------------------------------------------
__global__ void __launch_bounds__(256) memwrite_kernel(
    const float* __restrict__ mem, const float* __restrict__ wt,
    const float* __restrict__ e_ws, const float* __restrict__ a_ws,
    float* __restrict__ mem_out) {
  const int b = blockIdx.x / M_;
  const int m = blockIdx.x % M_;
  const int tid = threadIdx.x;

  const float e0 = e_ws[(b * H_ + 0) * M_ + m];
  const float e1 = e_ws[(b * H_ + 1) * M_ + m];
  const float e2 = e_ws[(b * H_ + 2) * M_ + m];
  const float e3 = e_ws[(b * H_ + 3) * M_ + m];
  const float a0 = a_ws[(b * H_ + 0) * M_ + m];
  const float a1 = a_ws[(b * H_ + 1) * M_ + m];
  const float a2 = a_ws[(b * H_ + 2) * M_ + m];
  const float a3 = a_ws[(b * H_ + 3) * M_ + m];

  const float4* mr = (const float4*)(mem + ((size_t)b * M_ + m) * N_);
  float4* mo = (float4*)(mem_out + ((size_t)b * M_ + m) * N_);
  const float4* w0 = (const float4*)(wt + (size_t)(b * H_ + 0) * N_);
  const float4* w1 = w0 + (N_ / 4);
  const float4* w2 = w1 + (N_ / 4);
  const float4* w3 = w2 + (N_ / 4);

  for (int i = tid; i < N_ / 4; i += 256) {
    __builtin_prefetch(&mr[i + 512], 0, 0);   // look ahead 2 strides
    const float4 mv = mr[i];
    const float4 v0 = w0[i], v1 = w1[i], v2 = w2[i], v3 = w3[i];
    float4 out;
#define DWM_COMP(c)                                                          \
    {                                                                        \
      const float kp = (1.f - e0 * v0.c) * (1.f - e1 * v1.c) *               \
                       (1.f - e2 * v2.c) * (1.f - e3 * v3.c);                \
      const float ad = a0 * v0.c + a1 * v1.c + a2 * v2.c + a3 * v3.c;        \
      out.c = mv.c * kp + ad;                                                \
    }
    DWM_COMP(x) DWM_COMP(y) DWM_COMP(z) DWM_COMP(w)
#undef DWM_COMP
    mo[i] = out;
  }
}

// ---------------------------------------------------------------------------
extern "C" void kernel_launch(void* const* d_in, const int* in_sizes, int n_in,
                              void* d_out, int out_size, void* d_ws, size_t ws_size,
                              hipStream_t stream) {
  (void)in_sizes; (void)n_in; (void)out_size; (void)ws_size;

  const float* x   = (const float*)d_in[0];
  const float* hid = (const float*)d_in[1];
  const float* hw  = (const float*)d_in[2];
  const float* sw  = (const float*)d_in[3];
  const float* mem = (const float*)d_in[4];
  const float* W_h = (const float*)d_in[5];
  const float* b_h = (const float*)d_in[6];
  const float* W_y = (const float*)d_in[7];
  const float* b_y = (const float*)d_in[8];
  const float* W_u = (const float*)d_in[9];
  const float* b_u = (const float*)d_in[10];

  // Outputs concatenated flat in return order: y, h_new, wt, snap_new, mem_new
  float* out      = (float*)d_out;
  float* y_out    = out;                                     // B*OUT   = 4096
  float* h_out    = y_out + B_ * OUT_;                       // B*STATE = 32768
  float* wt_out   = h_out + B_ * STATE;                      // B*H*N   = 2097152
  float* snap_out = wt_out + (size_t)B_ * H_ * N_;           // B*H*N
  float* mem_out  = snap_out + (size_t)B_ * H_ * N_;         // B*M*N

  float* ws       = (float*)d_ws;
  float* comb_ws  = ws;                                      // B*COMB   = 45056
  float* u_ws     = comb_ws + B_ * COMB_;                    // B*UPDATE = 18432
  float* e_ws     = u_ws + B_ * UPDATE_;                     // B*H*M    = 8192
  float* a_ws     = e_ws + B_ * H_ * M_;                     // B*H*M    = 8192

  pack_kernel<<<dim3(B_), 256, 0, stream>>>(x, hid, comb_ws);
  read_kernel<<<dim3(B_ * M_), 256, 0, stream>>>(mem, hw, comb_ws);

  controller_kernel<STATE, 0><<<dim3(4, STATE / 16), 32, 0, stream>>>(
      comb_ws, W_h, b_h, h_out);
  controller_kernel<OUT_, 1><<<dim3(4, OUT_ / 16), 32, 0, stream>>>(
      comb_ws, W_y, b_y, y_out);
  controller_kernel<UPDATE_, 2><<<dim3(4, UPDATE_ / 16), 32, 0, stream>>>(
      comb_ws, W_u, b_u, u_ws);

  address_kernel<<<dim3(B_ * H_), 256, 0, stream>>>(
      hw, sw, u_ws, wt_out, snap_out, e_ws, a_ws);
  memwrite_kernel<<<dim3(B_ * M_), 256, 0, stream>>>(
      mem, wt_out, e_ws, a_ws, mem_out);
}